// TrilinearInterpolate_26225070309540
// MI455X (gfx1250) — compile-verified
//
#include <hip/hip_runtime.h>
#include <hip/hip_bf16.h>

typedef __attribute__((ext_vector_type(2))) float v2f;
typedef __attribute__((ext_vector_type(8))) float v8f;

#define B_  4
#define N_  16384
#define M_  4096
#define C_  256
#define EPSW 1e-8f

// ---------------------------------------------------------------------------
// Kernel 0: prep known points into WMMA-A-friendly records:
//   kprep[g] = { -2x, -2y, -2z, x^2+y^2+z^2 }
// ---------------------------------------------------------------------------
__global__ void __launch_bounds__(256)
tn_prep_known(const float* __restrict__ known, float* __restrict__ kprep, int total) {
    int g = blockIdx.x * blockDim.x + threadIdx.x;
    if (g >= total) return;
    float x = known[3 * g + 0];
    float y = known[3 * g + 1];
    float z = known[3 * g + 2];
    float4 r;
    r.x = -2.0f * x;
    r.y = -2.0f * y;
    r.z = -2.0f * z;
    r.w = x * x + y * y + z * z;
    ((float4*)kprep)[g] = r;
}

// ---------------------------------------------------------------------------
// Kernel 1: 3-NN via V_WMMA_F32_16X16X4_F32.
//   D[mknown, nquery] = -2*(u.k) + ||k||^2   (||u||^2 added at the end)
// One wave handles 16 queries; block = 8 waves = 128 queries; known table
// for the batch (64KB) staged in LDS once per block. Inner loop is unrolled
// x4: 4 ds_load_b64 + 4 back-to-back WMMAs per group, one reject branch
// covering 64 candidates.
// ---------------------------------------------------------------------------
__device__ __forceinline__ void tn_insert3(float d, int mi,
                                           float& d0, float& d1, float& d2,
                                           int& j0, int& j1, int& j2) {
    bool lt2 = d < d2, lt1 = d < d1, lt0 = d < d0;
    float nd2 = lt1 ? d1 : (lt2 ? d  : d2);
    int   nj2 = lt1 ? j1 : (lt2 ? mi : j2);
    float nd1 = lt0 ? d0 : (lt1 ? d  : d1);
    int   nj1 = lt0 ? j0 : (lt1 ? mi : j1);
    d0 = lt0 ? d  : d0;
    j0 = lt0 ? mi : j0;
    d1 = nd1; j1 = nj1;
    d2 = nd2; j2 = nj2;
}

__global__ void __launch_bounds__(256)
tn_three_nn_wmma(const float* __restrict__ unknown,
                 const float* __restrict__ kprep,
                 int*   __restrict__ idx_out,
                 float* __restrict__ w_out) {
    __shared__ float lds_k[M_ * 4];  // 64KB: {-2x,-2y,-2z,ksq} per known point

    const int b    = blockIdx.x / (N_ / 128);
    const int qblk = (blockIdx.x % (N_ / 128)) * 128;
    const int tid  = threadIdx.x;

    // ---- stage known records for this batch into LDS (coalesced b128) ----
    {
        const float4* src = (const float4*)(kprep + (size_t)b * M_ * 4);
        float4*       dst = (float4*)lds_k;
        for (int r = tid; r < M_; r += 256) dst[r] = src[r];
    }
    __syncthreads();

    const int wave = tid >> 5;
    const int lane = tid & 31;
    const int hi   = lane >> 4;   // 0: K=0,1 half / M=r ; 1: K=2,3 half / M=8+r
    const int l16  = lane & 15;
    const int q    = qblk + wave * 16 + l16;   // query owned by this lane (N=l16)

    // ---- B operand (queries): lanes 0-15 -> (ux,uy)=K0,K1 ; 16-31 -> (uz,1)=K2,K3
    const float* up = unknown + ((size_t)b * N_ + q) * 3;
    const float ux = up[0], uy = up[1], uz = up[2];
    const float usq = ux * ux + uy * uy + uz * uz;
    v2f bmat;
    bmat.x = hi ? uz   : ux;
    bmat.y = hi ? 1.0f : uy;

    // running top-3 (ascending) per lane
    float d0 = 3.0e38f, d1 = 3.0e38f, d2 = 3.0e38f;
    int   j0 = 0,       j1 = 0,       j2 = 0;

    // lane-fixed LDS base; each 16-point tile is 64 floats (256B) apart
    const float* lbase = &lds_k[l16 * 4 + hi * 2];

    for (int mt = 0; mt < M_ / 16; mt += 4) {
        // ---- 4 A operands: ds_load_b64 with immediate offsets off one base
        const float* lp = lbase + mt * 64;
        v2f a0 = *(const v2f*)(lp + 0 * 64);
        v2f a1 = *(const v2f*)(lp + 1 * 64);
        v2f a2 = *(const v2f*)(lp + 2 * 64);
        v2f a3 = *(const v2f*)(lp + 3 * 64);

        // ---- 4 independent WMMAs, back-to-back (pipeline the XDL) ----
        v8f c0 = {}, c1 = {}, c2 = {}, c3 = {};
        c0 = __builtin_amdgcn_wmma_f32_16x16x4_f32(false, a0, false, bmat, (short)0, c0, false, false);
        c1 = __builtin_amdgcn_wmma_f32_16x16x4_f32(false, a1, false, bmat, (short)0, c1, false, false);
        c2 = __builtin_amdgcn_wmma_f32_16x16x4_f32(false, a2, false, bmat, (short)0, c2, false, false);
        c3 = __builtin_amdgcn_wmma_f32_16x16x4_f32(false, a3, false, bmat, (short)0, c3, false, false);

        // ---- fast reject over all 64 candidates (min3 tree) ----
        float m0 = fminf(fminf(fminf(c0[0], c0[1]), fminf(c0[2], c0[3])),
                         fminf(fminf(c0[4], c0[5]), fminf(c0[6], c0[7])));
        float m1 = fminf(fminf(fminf(c1[0], c1[1]), fminf(c1[2], c1[3])),
                         fminf(fminf(c1[4], c1[5]), fminf(c1[6], c1[7])));
        float m2 = fminf(fminf(fminf(c2[0], c2[1]), fminf(c2[2], c2[3])),
                         fminf(fminf(c2[4], c2[5]), fminf(c2[6], c2[7])));
        float m3 = fminf(fminf(fminf(c3[0], c3[1]), fminf(c3[2], c3[3])),
                         fminf(fminf(c3[4], c3[5]), fminf(c3[6], c3[7])));
        float mn = fminf(fminf(m0, m1), fminf(m2, m3));

        if (mn < d2) {   // rare: O(K log m) triggers per query over whole scan
            const int mb = mt * 16 + hi * 8;
#pragma unroll
            for (int r = 0; r < 8; ++r) tn_insert3(c0[r], mb +  0 + r, d0, d1, d2, j0, j1, j2);
#pragma unroll
            for (int r = 0; r < 8; ++r) tn_insert3(c1[r], mb + 16 + r, d0, d1, d2, j0, j1, j2);
#pragma unroll
            for (int r = 0; r < 8; ++r) tn_insert3(c2[r], mb + 32 + r, d0, d1, d2, j0, j1, j2);
#pragma unroll
            for (int r = 0; r < 8; ++r) tn_insert3(c3[r], mb + 48 + r, d0, d1, d2, j0, j1, j2);
        }
    }

    // ---- merge the two half-rows (lane q and lane q+16 hold same query) ----
    float od[3] = { d0, d1, d2 };
    int   oj[3] = { j0, j1, j2 };
#pragma unroll
    for (int t = 0; t < 3; ++t) {
        float d  = __shfl_xor(od[t], 16);
        int   mi = __shfl_xor(oj[t], 16);
        tn_insert3(d, mi, d0, d1, d2, j0, j1, j2);
    }

    if (hi == 0) {
        // restore true squared distances, then inverse-distance weights
        float s0 = sqrtf(fmaxf(d0 + usq, 0.0f));
        float s1 = sqrtf(fmaxf(d1 + usq, 0.0f));
        float s2 = sqrtf(fmaxf(d2 + usq, 0.0f));
        float r0 = 1.0f / (s0 + EPSW);
        float r1 = 1.0f / (s1 + EPSW);
        float r2 = 1.0f / (s2 + EPSW);
        float rs = 1.0f / (r0 + r1 + r2);
        size_t o = ((size_t)b * N_ + q) * 3;
        w_out[o + 0] = r0 * rs;
        w_out[o + 1] = r1 * rs;
        w_out[o + 2] = r2 * rs;
        idx_out[o + 0] = j0;
        idx_out[o + 1] = j1;
        idx_out[o + 2] = j2;
    }
}

// ---------------------------------------------------------------------------
// Kernel 2: feature gather + blend.  out[b][c][i] = sum_k w_k * F[b][c][idx_k]
// Thread-per-query (coalesced writes along n), loop over C; feats rows are
// L2-resident (16MB total vs 192MB L2).
// ---------------------------------------------------------------------------
__global__ void __launch_bounds__(256)
tn_three_interp(const float* __restrict__ feats,
                const int*   __restrict__ idx,
                const float* __restrict__ w,
                float* __restrict__ out) {
    const int b = blockIdx.x / (N_ / 256);
    const int i = (blockIdx.x % (N_ / 256)) * 256 + threadIdx.x;
    const size_t p = ((size_t)b * N_ + i) * 3;
    const int   i0 = idx[p + 0], i1 = idx[p + 1], i2 = idx[p + 2];
    const float w0 = w[p + 0],  w1 = w[p + 1],  w2 = w[p + 2];
    const float* F = feats + (size_t)b * C_ * M_;
    float*       O = out   + (size_t)b * C_ * N_ + i;
#pragma unroll 4
    for (int c = 0; c < C_; ++c) {
        const float* Fr = F + (size_t)c * M_;
        O[(size_t)c * N_] = w0 * Fr[i0] + w1 * Fr[i1] + w2 * Fr[i2];
    }
}

// ---------------------------------------------------------------------------
extern "C" void kernel_launch(void* const* d_in, const int* in_sizes, int n_in,
                              void* d_out, int out_size, void* d_ws, size_t ws_size,
                              hipStream_t stream) {
    const float* unknown = (const float*)d_in[0];   // (B, n, 3)
    const float* known   = (const float*)d_in[1];   // (B, m, 3)
    const float* feats   = (const float*)d_in[2];   // (B, C, m)
    float* out = (float*)d_out;                     // (B, C, n)

    // workspace carve-up
    float* kprep = (float*)d_ws;                                        // B*M*4 f32
    char*  p1    = (char*)d_ws + (size_t)B_ * M_ * 4 * sizeof(float);
    int*   nn_i  = (int*)p1;                                            // B*N*3 i32
    float* nn_w  = (float*)(p1 + (size_t)B_ * N_ * 3 * sizeof(int));    // B*N*3 f32

    tn_prep_known<<<(B_ * M_ + 255) / 256, 256, 0, stream>>>(known, kprep, B_ * M_);
    tn_three_nn_wmma<<<B_ * (N_ / 128), 256, 0, stream>>>(unknown, kprep, nn_i, nn_w);
    tn_three_interp<<<B_ * (N_ / 256), 256, 0, stream>>>(feats, nn_i, nn_w, out);
}